// LearnableHarmonicSynth_37873021616821
// MI455X (gfx1250) — compile-verified
//
#include <hip/hip_runtime.h>
#include <hip/hip_bf16.h>

typedef float v2f __attribute__((ext_vector_type(2)));
typedef float v8f __attribute__((ext_vector_type(8)));

#define SRF 48000.0f
#define TWO_PI_APPROX 6.28318f /* 2 * 3.14159 */

// out[i] = ( sum_{m=1..8} c_m * sin(m * x_i + phase) ) * inv_hsum
// x_i = f*sr*t0 + i * f*L/(L-1)
// Harmonic accumulation done on the matrix pipe with V_WMMA_F32_16X16X4_F32:
//   A[m][k] = c_k (broadcast over rows)  -> result row-independent
//   B[k][n] = sin((kbase+k+1) * x_n + ph) for 16 samples n
// Two chained K=4 WMMAs cover multipliers 1..8.
__global__ __launch_bounds__(256) void synth_wmma_kernel(
    const float* __restrict__ freq,
    const float* __restrict__ hamps,
    const float* __restrict__ t,
    const float* __restrict__ phase,
    float* __restrict__ out,
    int length)
{
    const float f      = freq[0];
    const float t0     = t[0];
    const float ph     = phase[0];
    const float scale0 = hamps[0];

    // normalization: sum |hamps[1..8]|
    float hsum = 0.0f;
#pragma unroll
    for (int j = 1; j <= 8; ++j) hsum += fabsf(hamps[j]);
    const float inv = 1.0f / hsum;

    // coefficients c[m-1] for multiplier m = 1..8
    const float freq_hz = f * SRF / TWO_PI_APPROX;
    float c[8];
    c[0] = scale0 * inv;
#pragma unroll
    for (int m = 2; m <= 8; ++m) {
        const float sel = (freq_hz * (float)m > SRF * 0.5f) ? 1.0e-4f : scale0;
        c[m - 1] = sel * hamps[m] * inv;
    }

    // x_i = x0 + i*dx  (double to keep phase accuracy over 8.4M samples)
    const double x0 = (double)f * (double)SRF * (double)t0;
    const double dx = (double)f * ((double)length / (double)(length - 1));

    const unsigned lane  = threadIdx.x & 31u;
    const unsigned n     = lane & 15u;   // sample column within the 16-wide group
    const unsigned khalf = lane >> 4;    // 0: K={0,1}, 1: K={2,3} (A/B VGPR layout)

    // A operands (coefficients, broadcast across the M dimension)
    v2f a0, a1;
    a0.x = khalf ? c[2] : c[0];
    a0.y = khalf ? c[3] : c[1];
    a1.x = khalf ? c[6] : c[4];
    a1.y = khalf ? c[7] : c[5];

    // This lane's harmonic multipliers for the two K-chunks
    const float m00 = (float)(2u * khalf + 1u);
    const float m01 = (float)(2u * khalf + 2u);
    const float m10 = (float)(2u * khalf + 5u);
    const float m11 = (float)(2u * khalf + 6u);

    const unsigned waveId  = (blockIdx.x * blockDim.x + threadIdx.x) >> 5;
    const unsigned nWaves  = (gridDim.x * blockDim.x) >> 5;
    const unsigned nChunks = ((unsigned)length + 31u) >> 5;  // 32 samples / chunk

    for (unsigned chk = waveId; chk < nChunks; chk += nWaves) {
        const unsigned base = chk * 32u;

        // group A: samples base+0..15, group B: samples base+16..31
        const float xA = (float)(x0 + (double)(base + n)        * dx);
        const float xB = (float)(x0 + (double)(base + 16u + n)  * dx);

        v2f bA0, bA1, bB0, bB1;
        bA0.x = __sinf(__builtin_fmaf(m00, xA, ph));
        bA0.y = __sinf(__builtin_fmaf(m01, xA, ph));
        bA1.x = __sinf(__builtin_fmaf(m10, xA, ph));
        bA1.y = __sinf(__builtin_fmaf(m11, xA, ph));
        bB0.x = __sinf(__builtin_fmaf(m00, xB, ph));
        bB0.y = __sinf(__builtin_fmaf(m01, xB, ph));
        bB1.x = __sinf(__builtin_fmaf(m10, xB, ph));
        bB1.y = __sinf(__builtin_fmaf(m11, xB, ph));

        v8f accA = {0.f, 0.f, 0.f, 0.f, 0.f, 0.f, 0.f, 0.f};
        v8f accB = {0.f, 0.f, 0.f, 0.f, 0.f, 0.f, 0.f, 0.f};

        accA = __builtin_amdgcn_wmma_f32_16x16x4_f32(false, a0, false, bA0,
                                                     (short)0, accA, false, false);
        accA = __builtin_amdgcn_wmma_f32_16x16x4_f32(false, a1, false, bA1,
                                                     (short)0, accA, false, false);
        accB = __builtin_amdgcn_wmma_f32_16x16x4_f32(false, a0, false, bB0,
                                                     (short)0, accB, false, false);
        accB = __builtin_amdgcn_wmma_f32_16x16x4_f32(false, a1, false, bB1,
                                                     (short)0, accB, false, false);

        // D row 0 (lanes 0-15) and row 8 (lanes 16-31) both carry the dot
        // product for column n of their group -> every lane holds sample
        // base+lane's value after this select. 128B coalesced NT store.
        const float v = (khalf == 0u) ? accA[0] : accB[0];
        const unsigned idx = base + lane;
        if (idx < (unsigned)length)
            __builtin_nontemporal_store(v, out + idx);
    }
}

extern "C" void kernel_launch(void* const* d_in, const int* in_sizes, int n_in,
                              void* d_out, int out_size, void* d_ws, size_t ws_size,
                              hipStream_t stream) {
    // setup_inputs order:
    // 0 feedback_line, 1 freq, 2 output_length_samples, 3 hamps, 4 t, 5 phase,
    // 6 W1, 7 b1, 8 W2, 9 b2, 10 W3, 11 b3   (MLP is dead code in the reference)
    const float* freq  = (const float*)d_in[1];
    const float* hamps = (const float*)d_in[3];
    const float* t     = (const float*)d_in[4];
    const float* phase = (const float*)d_in[5];
    float* out = (float*)d_out;

    const int length = out_size;  // == output_length_samples (8388608)

    const int block = 256;        // 8 waves per workgroup (wave32)
    const int grid  = 2048;       // 16384 waves; 262144 chunks -> 16 chunks/wave

    synth_wmma_kernel<<<grid, block, 0, stream>>>(freq, hamps, t, phase, out, length);
}